// EMA_78683800862946
// MI455X (gfx1250) — compile-verified
//
#include <hip/hip_runtime.h>
#include <math.h>

typedef float v2f __attribute__((ext_vector_type(2)));
typedef float v8f __attribute__((ext_vector_type(8)));

// ---------------- workspace layout (in floats) ----------------
// [0,    4096)  Mtab[l][i][j]  (16*16*16)
// [4096, 4352)  vtab[l][i] = a^(i+1)
// [4352, 4368)  atab[l]
// [4368, 4384)  altab[l] (alpha)
// [4384, 4400)  a64tab[l] = a^64
// [4400, 5424)  init[n]  (1024)
// [8192, 8192+64*16*1024)  bbuf: K2 partials -> K3 rewrites to pre-chunk carries
#define WS_MTAB   0
#define WS_VTAB   4096
#define WS_A      4352
#define WS_AL     4368
#define WS_A64    4384
#define WS_INIT   4400
#define WS_BBUF   8192

#define NLAY 16
#define NCOL 1024          // B*D
#define TLEN 4096
#define CH   64            // chunk length (timesteps)
#define NCHUNK (TLEN/CH)   // 64

// K0: per-layer tables
__global__ void ema_k0_tables(const float* __restrict__ log_decay, float* __restrict__ ws) {
    int tid = threadIdx.x;            // 256 threads: (l, i)
    int l = tid >> 4, i = tid & 15;
    float ld    = log_decay[l];
    float sg    = 1.0f / (1.0f + expf(-ld));
    float decay = sg * sg * 510.0f + 2.0f;
    float alpha = 1.0f / decay;
    float a     = 1.0f - alpha;
    // M row i: M[i][j] = alpha * a^(i-j) for j<=i else 0
    float* Mrow = ws + WS_MTAB + (l * 16 + i) * 16;
    float p = 1.0f;
    for (int j = i; j >= 0; --j) { Mrow[j] = alpha * p; p *= a; }
    for (int j = i + 1; j < 16; ++j) Mrow[j] = 0.0f;
    // v[i] = a^(i+1)
    float pv = a;
    for (int k = 0; k < i; ++k) pv *= a;
    ws[WS_VTAB + l * 16 + i] = pv;
    if (i == 0) {
        ws[WS_A  + l] = a;
        ws[WS_AL + l] = alpha;
        float p64 = 1.0f;
        for (int k = 0; k < CH; ++k) p64 *= a;
        ws[WS_A64 + l] = p64;
    }
}

// K1: init[n] = mean of x[0:256, n]
__global__ void ema_k1_init(const float* __restrict__ x, float* __restrict__ ws) {
    int n = blockIdx.x * blockDim.x + threadIdx.x;   // 1024 threads
    float s = 0.0f;
    for (int t = 0; t < 256; ++t) s += x[t * NCOL + n];
    ws[WS_INIT + n] = s * (1.0f / 256.0f);
}

// K2: per-chunk partial B_c[l][n] = scan of 64 steps from zero state
__global__ void ema_k2_partial(const float* __restrict__ x, float* __restrict__ ws) {
    int tid = blockIdx.x * blockDim.x + threadIdx.x; // 65536 threads: (c, n)
    int c = tid >> 10, n = tid & (NCOL - 1);
    float a[NLAY], al[NLAY], s[NLAY];
#pragma unroll
    for (int l = 0; l < NLAY; ++l) {
        a[l]  = ws[WS_A  + l];
        al[l] = ws[WS_AL + l];
        s[l]  = 0.0f;
    }
    const float* xp = x + (size_t)c * CH * NCOL + n;
    for (int j = 0; j < CH; ++j) {
        float xv = xp[(size_t)j * NCOL];
#pragma unroll
        for (int l = 0; l < NLAY; ++l) s[l] = fmaf(a[l], s[l], al[l] * xv);
    }
    float* bbuf = ws + WS_BBUF;
#pragma unroll
    for (int l = 0; l < NLAY; ++l) bbuf[((size_t)c * NLAY + l) * NCOL + n] = s[l];
}

// K3: serial prefix over chunks; rewrite bbuf[c] with state BEFORE chunk c
__global__ void ema_k3_prefix(float* __restrict__ ws) {
    int tid = blockIdx.x * blockDim.x + threadIdx.x; // 16384 threads: (l, n)
    int l = tid >> 10, n = tid & (NCOL - 1);
    float y   = ws[WS_INIT + n];
    float a64 = ws[WS_A64 + l];
    float* bbuf = ws + WS_BBUF;
    for (int c = 0; c < NCHUNK; ++c) {
        float* p = bbuf + ((size_t)c * NLAY + l) * NCOL + n;
        float b = *p;
        *p = y;                 // carry entering chunk c
        y = fmaf(a64, y, b);
    }
}

// K4: WMMA output phase. Block = 512 threads = 16 waves (one per layer).
// blockIdx: (chunk c, 16-column group). Each wave: 4 sub-tiles of chained
// V_WMMA_F32_16X16X4_F32; carry propagated inside the wave via shfl of D row 15.
__global__ void __launch_bounds__(512) ema_k4_wmma(const float* __restrict__ x,
                                                   const float* __restrict__ ws,
                                                   float* __restrict__ out) {
    __shared__ float lds[16 * 257];   // [layer][t*16+n] padded stride 257
    const float* Mtab  = ws + WS_MTAB;
    const float* vtab  = ws + WS_VTAB;
    const float* carry = ws + WS_BBUF;

    int c    = blockIdx.x >> 6;        // 64 chunks
    int nb   = (blockIdx.x & 63) << 4; // column group base (64 groups of 16)
    int l    = threadIdx.x >> 5;       // wave id == layer
    int lane = threadIdx.x & 31;
    int half = lane >> 4;
    int mrow = lane & 15;              // A row / B+D column within tile

    // A fragments (constant per layer): slice q covers K = 4q..4q+3;
    // this lane holds local K = 2*half + {0,1}
    v2f afrag[4];
#pragma unroll
    for (int q = 0; q < 4; ++q) {
        const float* mp = Mtab + (l * 16 + mrow) * 16 + (4 * q + 2 * half);
        afrag[q] = v2f{mp[0], mp[1]};
    }
    float vt[8];
#pragma unroll
    for (int r = 0; r < 8; ++r) vt[r] = vtab[l * 16 + 8 * half + r];

    // carry entering this 64-step chunk
    float ycar = carry[((size_t)c * NLAY + l) * NCOL + nb + mrow];

    const long tbase = (long)c * CH;
    for (int sub = 0; sub < 4; ++sub) {
        const long t0 = tbase + sub * 16;
        // B fragments: X[j][n], j = 4q + 2*half + {0,1}, n = nb + mrow
        v2f bfrag[4];
#pragma unroll
        for (int q = 0; q < 4; ++q) {
            const float* xp = x + (t0 + 4 * q + 2 * half) * NCOL + nb + mrow;
            bfrag[q] = v2f{xp[0], xp[NCOL]};
        }
        // C init with carry term: row i = r + 8*half -> v[i] * ycar
        v8f d;
#pragma unroll
        for (int r = 0; r < 8; ++r) d[r] = vt[r] * ycar;
        // D = M * X + C, chained over 4 K-slices
#pragma unroll
        for (int q = 0; q < 4; ++q)
            d = __builtin_amdgcn_wmma_f32_16x16x4_f32(
                    false, afrag[q], false, bfrag[q], (short)0, d, false, false);
        // carry for next sub-tile: y_end = D[15][n], held by lane 16+n in vgpr 7
        ycar = __shfl(d[7], 16 + mrow, 32);
        // stage D tile to LDS
#pragma unroll
        for (int r = 0; r < 8; ++r)
            lds[l * 257 + (8 * half + r) * 16 + mrow] = d[r];
        __syncthreads();
        // cooperative coalesced writeout: 16 t x (16 n x 16 l) contiguous runs
        const long obase = t0 * (long)(NCOL * NLAY) + (long)nb * NLAY;
#pragma unroll
        for (int k = 0; k < 8; ++k) {
            int f  = threadIdx.x + 512 * k;      // 4096 floats per sub-tile
            int ll = f & 15;
            int nn = (f >> 4) & 15;
            int tt = f >> 8;
            out[obase + (long)tt * (NCOL * NLAY) + nn * NLAY + ll] =
                lds[ll * 257 + tt * 16 + nn];
        }
        __syncthreads();
    }
}

extern "C" void kernel_launch(void* const* d_in, const int* in_sizes, int n_in,
                              void* d_out, int out_size, void* d_ws, size_t ws_size,
                              hipStream_t stream) {
    const float* x         = (const float*)d_in[0];   // [4096, 4, 256]
    const float* log_decay = (const float*)d_in[1];   // [16]
    float* out = (float*)d_out;                        // [4096, 4, 256, 16]
    float* ws  = (float*)d_ws;                         // needs ~4.1 MB

    ema_k0_tables <<<1,    256, 0, stream>>>(log_decay, ws);
    ema_k1_init   <<<4,    256, 0, stream>>>(x, ws);
    ema_k2_partial<<<256,  256, 0, stream>>>(x, ws);
    ema_k3_prefix <<<64,   256, 0, stream>>>(ws);
    ema_k4_wmma   <<<4096, 512, 0, stream>>>(x, ws, out);
    (void)in_sizes; (void)n_in; (void)out_size; (void)ws_size;
}